// AttnDecoderRNN_19524921327937
// MI455X (gfx1250) — compile-verified
//
#include <hip/hip_runtime.h>
#include <math.h>

typedef float v2f __attribute__((ext_vector_type(2)));
typedef float v8f __attribute__((ext_vector_type(8)));

#define NB   64
#define SEQ  256
#define HD   1024
#define HD2  2048
#define HD3  3072

// ---- CDNA5 f32 WMMA: D(16x16) = A(16x4) * B(4x16) + C ----------------------
// A layout (ISA 7.12.2): lanes 0-15 hold (M=lane, K=0..1), lanes 16-31 hold
// (M=lane-16, K=2..3) in 2 VGPRs. B mirrors with N in place of M.
// C/D: VGPR e = row (e + 8*(lane>=16)), col = lane&15.
static __device__ __forceinline__ v8f wmma_f32_16x16x4(v2f a, v2f b, v8f c) {
    return __builtin_amdgcn_wmma_f32_16x16x4_f32(
        /*neg_a=*/false, a, /*neg_b=*/false, b,
        /*c_mod=*/(short)0, c, /*reuse_a=*/false, /*reuse_b=*/false);
}

static __device__ __forceinline__ float sigmoidf_(float x) {
    return 1.0f / (1.0f + __expf(-x));
}

// ---- K1: fused GRU cell ----------------------------------------------------
// One wave per (m_tile, j_tile): 16 batch rows x 16 hidden features.
// 6 f32 accumulators (r/z/n gates for gi and gh), K-loops over 2048 / 1024.
__global__ __launch_bounds__(128) void gru_kernel(
    const float* __restrict__ last_output, const float* __restrict__ last_context,
    const float* __restrict__ last_hidden, const float* __restrict__ W_ih,
    const float* __restrict__ b_ih, const float* __restrict__ W_hh,
    const float* __restrict__ b_hh, float* __restrict__ hidden_out)
{
    const int wave  = blockIdx.x * 4 + (threadIdx.x >> 5);
    const int lane  = threadIdx.x & 31;
    const int m0    = (wave >> 6) * 16;   // batch tile (0..3)
    const int j0    = (wave & 63) * 16;   // hidden-feature tile (0..63)
    const int row   = lane & 15;
    const int khalf = lane >> 4;

    v8f acc_ir = {}, acc_iz = {}, acc_in = {};
    v8f acc_hr = {}, acc_hz = {}, acc_hn = {};

    const int jr = j0 + row;
    const float* wih_r = W_ih + (size_t)(jr)          * HD2;
    const float* wih_z = W_ih + (size_t)(jr + HD)     * HD2;
    const float* wih_n = W_ih + (size_t)(jr + 2 * HD) * HD2;
    const float* whh_r = W_hh + (size_t)(jr)          * HD;
    const float* whh_z = W_hh + (size_t)(jr + HD)     * HD;
    const float* whh_n = W_hh + (size_t)(jr + 2 * HD) * HD;
    const int am = m0 + row;

    // gi = [last_output | last_context] @ W_ih^T   (K = 2048)
    for (int k = 0; k < HD2; k += 4) {
        const int kk = k + khalf * 2;
        v2f a = (kk < HD)
              ? *(const v2f*)(last_output  + (size_t)am * HD + kk)
              : *(const v2f*)(last_context + (size_t)am * HD + (kk - HD));
        v2f br = *(const v2f*)(wih_r + kk);
        v2f bz = *(const v2f*)(wih_z + kk);
        v2f bn = *(const v2f*)(wih_n + kk);
        acc_ir = wmma_f32_16x16x4(a, br, acc_ir);
        acc_iz = wmma_f32_16x16x4(a, bz, acc_iz);
        acc_in = wmma_f32_16x16x4(a, bn, acc_in);
    }
    // gh = h @ W_hh^T   (K = 1024)
    for (int k = 0; k < HD; k += 4) {
        const int kk = k + khalf * 2;
        v2f a  = *(const v2f*)(last_hidden + (size_t)am * HD + kk);
        v2f br = *(const v2f*)(whh_r + kk);
        v2f bz = *(const v2f*)(whh_z + kk);
        v2f bn = *(const v2f*)(whh_n + kk);
        acc_hr = wmma_f32_16x16x4(a, br, acc_hr);
        acc_hz = wmma_f32_16x16x4(a, bz, acc_hz);
        acc_hn = wmma_f32_16x16x4(a, bn, acc_hn);
    }

    const float bir = b_ih[jr], biz = b_ih[jr + HD], bin_ = b_ih[jr + 2 * HD];
    const float bhr = b_hh[jr], bhz = b_hh[jr + HD], bhn  = b_hh[jr + 2 * HD];

    #pragma unroll
    for (int e = 0; e < 8; ++e) {
        const int n = m0 + e + khalf * 8;   // batch row of C element e
        float ir = acc_ir[e] + bir, hr = acc_hr[e] + bhr;
        float iz = acc_iz[e] + biz, hz = acc_hz[e] + bhz;
        float in_ = acc_in[e] + bin_, hn = acc_hn[e] + bhn;
        float r  = sigmoidf_(ir + hr);
        float z  = sigmoidf_(iz + hz);
        float nn = tanhf(in_ + r * hn);
        float hp = last_hidden[(size_t)n * HD + jr];
        hidden_out[(size_t)n * HD + jr] = (1.0f - z) * nn + z * hp;
    }
}

// ---- K2: v = h_new @ W_attn  (64x1024 @ 1024x1024) -------------------------
__global__ __launch_bounds__(128) void attnq_kernel(
    const float* __restrict__ hidden, const float* __restrict__ W_attn,
    float* __restrict__ v_out)
{
    const int wave  = blockIdx.x * 4 + (threadIdx.x >> 5);
    const int lane  = threadIdx.x & 31;
    const int m0    = (wave >> 6) * 16;
    const int c0    = (wave & 63) * 16;
    const int row   = lane & 15;
    const int khalf = lane >> 4;
    const int am = m0 + row;
    const int cc = c0 + row;

    v8f acc = {};
    for (int k = 0; k < HD; k += 4) {
        const int kk = k + khalf * 2;
        v2f a = *(const v2f*)(hidden + (size_t)am * HD + kk);
        v2f b;
        b.x = W_attn[(size_t)kk * HD + cc];
        b.y = W_attn[(size_t)(kk + 1) * HD + cc];
        acc = wmma_f32_16x16x4(a, b, acc);
    }
    #pragma unroll
    for (int e = 0; e < 8; ++e) {
        const int n = m0 + e + khalf * 8;
        v_out[(size_t)n * HD + cc] = acc[e];
    }
}

// ---- K3: scores[n,s] = enc[n,s,:] . v[n,:]  (b_attn term cancels in softmax)
__global__ __launch_bounds__(256) void scores_kernel(
    const float* __restrict__ enc, const float* __restrict__ v,
    float* __restrict__ scores)
{
    const int g    = blockIdx.x * 8 + (threadIdx.x >> 5);  // one wave per (n,s)
    const int lane = threadIdx.x & 31;
    const int n = g >> 8;
    const int s = g & 255;
    const float4* erow = (const float4*)(enc + ((size_t)n * SEQ + s) * HD);
    const float4* vrow = (const float4*)(v + (size_t)n * HD);
    float sum = 0.f;
    #pragma unroll
    for (int i = 0; i < 8; ++i) {
        float4 e4 = erow[lane + 32 * i];
        float4 v4 = vrow[lane + 32 * i];
        sum += e4.x * v4.x + e4.y * v4.y + e4.z * v4.z + e4.w * v4.w;
    }
    #pragma unroll
    for (int off = 16; off > 0; off >>= 1)
        sum += __shfl_down(sum, off, 32);
    if (lane == 0) scores[(size_t)n * SEQ + s] = sum;
}

// ---- K4: row softmax over s (one block per n) ------------------------------
__global__ __launch_bounds__(256) void softmax_kernel(
    const float* __restrict__ scores, float* __restrict__ attn)
{
    __shared__ float red[SEQ];
    const int n = blockIdx.x, t = threadIdx.x;
    float x = scores[(size_t)n * SEQ + t];
    red[t] = x; __syncthreads();
    for (int off = 128; off > 0; off >>= 1) {
        if (t < off) red[t] = fmaxf(red[t], red[t + off]);
        __syncthreads();
    }
    const float m = red[0]; __syncthreads();
    const float e = __expf(x - m);
    red[t] = e; __syncthreads();
    for (int off = 128; off > 0; off >>= 1) {
        if (t < off) red[t] += red[t + off];
        __syncthreads();
    }
    attn[(size_t)n * SEQ + t] = e / red[0];
}

// ---- K5: context[n,h] = sum_s attn[n,s] * enc[n,s,h] (coalesced over h) ----
__global__ __launch_bounds__(256) void context_kernel(
    const float* __restrict__ enc, const float* __restrict__ attn,
    float* __restrict__ context)
{
    __shared__ float a[SEQ];
    const int n  = blockIdx.x >> 2;
    const int hc = blockIdx.x & 3;
    const int t  = threadIdx.x;
    a[t] = attn[(size_t)n * SEQ + t];
    __syncthreads();
    const int h = hc * 256 + t;
    const float* base = enc + (size_t)n * SEQ * HD + h;
    float acc = 0.f;
    #pragma unroll 4
    for (int s = 0; s < SEQ; ++s)
        acc += a[s] * base[(size_t)s * HD];
    context[(size_t)n * HD + h] = acc;
}

// ---- K6: output = [h_new | context] @ W_out^T + b_out ----------------------
__global__ __launch_bounds__(128) void out_kernel(
    const float* __restrict__ hidden, const float* __restrict__ context,
    const float* __restrict__ W_out, const float* __restrict__ b_out,
    float* __restrict__ out)
{
    const int wave  = blockIdx.x * 4 + (threadIdx.x >> 5);
    const int lane  = threadIdx.x & 31;
    const int m0    = (wave >> 6) * 16;
    const int j0    = (wave & 63) * 16;
    const int row   = lane & 15;
    const int khalf = lane >> 4;
    const int am = m0 + row, jr = j0 + row;
    const float* w = W_out + (size_t)jr * HD2;

    v8f acc = {};
    for (int k = 0; k < HD2; k += 4) {
        const int kk = k + khalf * 2;
        v2f a = (kk < HD)
              ? *(const v2f*)(hidden  + (size_t)am * HD + kk)
              : *(const v2f*)(context + (size_t)am * HD + (kk - HD));
        v2f b = *(const v2f*)(w + kk);
        acc = wmma_f32_16x16x4(a, b, acc);
    }
    const float bo = b_out[jr];
    #pragma unroll
    for (int e = 0; e < 8; ++e) {
        const int nb = m0 + e + khalf * 8;
        out[(size_t)nb * HD + jr] = acc[e] + bo;
    }
}

extern "C" void kernel_launch(void* const* d_in, const int* in_sizes, int n_in,
                              void* d_out, int out_size, void* d_ws, size_t ws_size,
                              hipStream_t stream) {
    const float* last_output  = (const float*)d_in[0];
    const float* last_context = (const float*)d_in[1];
    const float* last_hidden  = (const float*)d_in[2];
    const float* enc          = (const float*)d_in[3];
    const float* W_ih         = (const float*)d_in[4];
    const float* b_ih         = (const float*)d_in[5];
    const float* W_hh         = (const float*)d_in[6];
    const float* b_hh         = (const float*)d_in[7];
    const float* W_attn       = (const float*)d_in[8];
    // d_in[9] = b_attn: per-n constant in scores -> cancels in softmax, unused
    const float* W_out        = (const float*)d_in[10];
    const float* b_out        = (const float*)d_in[11];

    float* out_output  = (float*)d_out;                       // N*H
    float* out_context = (float*)d_out + NB * HD;             // N*H
    float* out_hidden  = (float*)d_out + 2 * NB * HD;         // N*H
    float* out_attn    = (float*)d_out + 3 * NB * HD;         // N*S

    float* ws_v      = (float*)d_ws;                          // N*H
    float* ws_scores = (float*)d_ws + NB * HD;                // N*S

    gru_kernel   <<<64,   128, 0, stream>>>(last_output, last_context, last_hidden,
                                            W_ih, b_ih, W_hh, b_hh, out_hidden);
    attnq_kernel <<<64,   128, 0, stream>>>(out_hidden, W_attn, ws_v);
    scores_kernel<<<2048, 256, 0, stream>>>(enc, ws_v, ws_scores);
    softmax_kernel<<<64,  256, 0, stream>>>(ws_scores, out_attn);
    context_kernel<<<256, 256, 0, stream>>>(enc, out_attn, out_context);
    out_kernel   <<<64,   128, 0, stream>>>(out_hidden, out_context, W_out, b_out, out_output);
}